// LengthAligner_83665962926843
// MI455X (gfx1250) — compile-verified
//
#include <hip/hip_runtime.h>
#include <math.h>

#define B_ 16
#define T_ 512
#define H_ 384
#define F_ 384
#define K_ 3
#define L_ 4096            // T_ * DMAX
#define C4_ (H_ / 4)       // 96 K-steps per tap
#define NT_ (F_ / 16)      // 24 N-tiles
#define SXS (H_ + 4)       // padded LDS row stride (bank-conflict-free A loads)
#define SYS (F_ + 4)       // padded LDS row stride for accumulator spill

typedef __attribute__((ext_vector_type(2))) float v2f;
typedef __attribute__((ext_vector_type(8))) float v8f;

// Repack torch OIW weights straight into WMMA B-fragment order:
//   wf[k][c4][nt][lane*2 + j] = w[f = nt*16 + (lane&15)][c = c4*4 + 2*(lane>>4) + j][k]
// One fragment = 64 consecutive floats = one coalesced global_load_b64 per lane.
__global__ void repack_w_kernel(const float* __restrict__ w, float* __restrict__ wf) {
    int idx = blockIdx.x * blockDim.x + threadIdx.x;
    const int n = K_ * C4_ * NT_ * 64;
    if (idx >= n) return;
    int j    = idx & 1;
    int lane = (idx >> 1) & 31;
    int nt   = (idx >> 6) % NT_;
    int c4   = (idx / (64 * NT_)) % C4_;
    int k    = idx / (64 * NT_ * C4_);
    int f = nt * 16 + (lane & 15);
    int c = c4 * 4 + 2 * (lane >> 4) + j;
    wf[idx] = w[(f * H_ + c) * K_ + k];
}

// Fused conv1d(K=3,same) -> +bias -> LayerNorm -> GELU(exact).
// Optionally fused linear head (F->1) + ReLU into len_out instead of writing hout.
// Grid: (B*T)/16 blocks, 256 threads (8 wave32). Each wave: 16x48 output tile,
// 3 f32 WMMA accumulators, K-loop = 3 taps x 384 channels, step 4.
__global__ __launch_bounds__(256) void conv_ln_gelu_kernel(
    const float* __restrict__ xin,   // [B,T,H]
    const float* __restrict__ wf,    // fragment-packed weights
    const float* __restrict__ bias,  // [F]
    const float* __restrict__ ln_g,  // [F]
    const float* __restrict__ ln_b,  // [F]
    const float* __restrict__ lin_w, // [F] or null
    const float* __restrict__ lin_b, // [1] or null
    float* __restrict__ hout,        // [B,T,F] or null
    float* __restrict__ len_out)     // [B,T] or null
{
    __shared__ float s_x[18 * SXS];  // halo'd token tile (padded rows)
    __shared__ float s_y[16 * SYS];  // pre-LN conv output tile (padded rows)

    const int tid = threadIdx.x;
    const int t0  = blockIdx.x * 16;
    const int b   = t0 / T_;
    const int tl  = t0 % T_;         // T_ % 16 == 0 -> tile never crosses batches
    const float* xb = xin + (size_t)b * T_ * H_;

    // stage tokens tl-1 .. tl+16 (zero outside batch)
    for (int i = tid; i < 18 * H_; i += 256) {
        int r = i / H_, c = i % H_;
        int tok = tl + r - 1;
        s_x[r * SXS + c] = (tok >= 0 && tok < T_) ? xb[tok * H_ + c] : 0.0f;
    }
    __syncthreads();

    const int wave = tid >> 5;
    const int lane = tid & 31;
    const int half = lane >> 4;      // K-half select per ISA A/B fragment layout
    const int l16  = lane & 15;
    const int fb   = wave * 48;

    v8f a0 = {}, a1 = {}, a2 = {};
    for (int k = 0; k < K_; ++k) {
        const float* xr = s_x + (l16 + k) * SXS;          // token row M=l16, tap k
        const float* wb = wf + ((k * C4_) * NT_ + wave * 3) * 64 + lane * 2;
        for (int c4 = 0; c4 < C4_; ++c4) {
            int ca = c4 * 4 + 2 * half;
            v2f av; av.x = xr[ca]; av.y = xr[ca + 1];
            const float* fr = wb + c4 * (NT_ * 64);
            v2f b0 = *(const v2f*)(fr);
            v2f b1 = *(const v2f*)(fr + 64);
            v2f b2 = *(const v2f*)(fr + 128);
            a0 = __builtin_amdgcn_wmma_f32_16x16x4_f32(false, av, false, b0, (short)0, a0, false, false);
            a1 = __builtin_amdgcn_wmma_f32_16x16x4_f32(false, av, false, b1, (short)0, a1, false, false);
            a2 = __builtin_amdgcn_wmma_f32_16x16x4_f32(false, av, false, b2, (short)0, a2, false, false);
        }
    }

    // spill D tiles (+bias) to LDS per ISA C/D layout: vgpr j -> row j + 8*half, col = l16
    float bi0 = bias[fb + l16], bi1 = bias[fb + 16 + l16], bi2 = bias[fb + 32 + l16];
    for (int j = 0; j < 8; ++j) {
        int row = j + 8 * half;
        s_y[row * SYS + fb + l16]      = a0[j] + bi0;
        s_y[row * SYS + fb + 16 + l16] = a1[j] + bi1;
        s_y[row * SYS + fb + 32 + l16] = a2[j] + bi2;
    }
    __syncthreads();

    // LayerNorm + GELU: 16 threads per token row
    const int row = tid >> 4;
    const int sl  = tid & 15;
    float vals[24];
    float s = 0.f, s2 = 0.f;
    for (int j = 0; j < 24; ++j) {
        float v = s_y[row * SYS + sl + 16 * j];
        vals[j] = v; s += v; s2 += v * v;
    }
    for (int off = 8; off >= 1; off >>= 1) {
        s  += __shfl_xor(s,  off, 16);
        s2 += __shfl_xor(s2, off, 16);
    }
    float mu  = s * (1.0f / F_);
    float var = s2 * (1.0f / F_) - mu * mu;
    float rs  = rsqrtf(var + 1e-5f);

    float dot = 0.f;
    for (int j = 0; j < 24; ++j) {
        int col = sl + 16 * j;
        float z  = (vals[j] - mu) * rs * ln_g[col] + ln_b[col];
        float ge = 0.5f * z * (1.0f + erff(z * 0.70710678118654752f));
        if (hout)    hout[((size_t)(b * T_ + tl + row)) * F_ + col] = ge;
        if (len_out) dot += ge * lin_w[col];
    }
    if (len_out) {
        for (int off = 8; off >= 1; off >>= 1) dot += __shfl_xor(dot, off, 16);
        if (sl == 0) {
            float v = dot + lin_b[0];
            len_out[b * T_ + tl + row] = v > 0.f ? v : 0.f;
        }
    }
}

// ends[b,t] = inclusive cumsum of round(exp(true_durations))
__global__ __launch_bounds__(512) void dur_scan_kernel(const float* __restrict__ td,
                                                       int* __restrict__ ends) {
    __shared__ int s[T_];
    int b = blockIdx.x, t = threadIdx.x;
    s[t] = (int)lrintf(expf(td[b * T_ + t]));   // round-half-even like jnp.round
    __syncthreads();
    for (int off = 1; off < T_; off <<= 1) {
        int v = (t >= off) ? s[t - off] : 0;
        __syncthreads();
        s[t] += v;
        __syncthreads();
    }
    ends[b * T_ + t] = s[t];
}

// out[b,l,:] = x[b, tok(l), :] for l < total, else 0
__global__ __launch_bounds__(256) void regulate_kernel(const float* __restrict__ x,
                                                       const int* __restrict__ ends,
                                                       float* __restrict__ out) {
    __shared__ int s_tok[8];
    __shared__ int s_tot;
    const int b  = blockIdx.y;
    const int l0 = blockIdx.x * 8;
    const int tid = threadIdx.x;
    const int* eb = ends + b * T_;
    if (tid < 8) {
        int l = l0 + tid;
        int lo = 0, hi = T_;
        while (lo < hi) { int mid = (lo + hi) >> 1; if (eb[mid] > l) hi = mid; else lo = mid + 1; }
        s_tok[tid] = lo;
    }
    if (tid == 0) s_tot = eb[T_ - 1];
    __syncthreads();
    const float4* x4 = (const float4*)(x + (size_t)b * T_ * H_);
    float4*       o4 = (float4*)(out + (size_t)b * L_ * H_);
    const int VR = H_ / 4;  // 96 float4 per row
    for (int i = tid; i < 8 * VR; i += 256) {
        int r = i / VR, e = i % VR;
        int l = l0 + r;
        float4 v;
        if (l < s_tot) v = x4[(size_t)s_tok[r] * VR + e];
        else           v = make_float4(0.f, 0.f, 0.f, 0.f);
        o4[(size_t)l * VR + e] = v;
    }
}

extern "C" void kernel_launch(void* const* d_in, const int* in_sizes, int n_in,
                              void* d_out, int out_size, void* d_ws, size_t ws_size,
                              hipStream_t stream) {
    const float* x   = (const float*)d_in[0];
    const float* td  = (const float*)d_in[1];
    const float* c1w = (const float*)d_in[2];
    const float* c1b = (const float*)d_in[3];
    const float* g1  = (const float*)d_in[4];
    const float* b1  = (const float*)d_in[5];
    const float* c2w = (const float*)d_in[6];
    const float* c2b = (const float*)d_in[7];
    const float* g2  = (const float*)d_in[8];
    const float* b2  = (const float*)d_in[9];
    const float* lw  = (const float*)d_in[10];
    const float* lb  = (const float*)d_in[11];

    // workspace layout
    const int nw = K_ * C4_ * NT_ * 64;              // == 3*384*384
    float* wf1 = (float*)d_ws;
    float* wf2 = wf1 + nw;
    float* h1  = wf2 + nw;                           // B*T*F
    int*   ends = (int*)(h1 + (size_t)B_ * T_ * F_); // B*T

    float* out = (float*)d_out;                      // [B,L,H]
    float* len = out + (size_t)B_ * L_ * H_;         // [B,T]

    repack_w_kernel<<<(nw + 255) / 256, 256, 0, stream>>>(c1w, wf1);
    repack_w_kernel<<<(nw + 255) / 256, 256, 0, stream>>>(c2w, wf2);

    conv_ln_gelu_kernel<<<(B_ * T_) / 16, 256, 0, stream>>>(
        x, wf1, c1b, g1, b1, nullptr, nullptr, h1, nullptr);
    conv_ln_gelu_kernel<<<(B_ * T_) / 16, 256, 0, stream>>>(
        h1, wf2, c2b, g2, b2, lw, lb, nullptr, len);

    dur_scan_kernel<<<B_, T_, 0, stream>>>(td, ends);
    regulate_kernel<<<dim3(L_ / 8, B_), 256, 0, stream>>>(x, ends, out);
}